// Hilbert_Sa_42245298324133
// MI455X (gfx1250) — compile-verified
//
#include <hip/hip_runtime.h>
#include <math.h>

// Problem constants (from reference: ORDER=8, POOL=8)
#define CHN   128      // 2*PSQ
#define PATHS 1024     // N_PIX / PSQ
#define NPIX  65536    // 256*256
#define PSQ   64

typedef __attribute__((ext_vector_type(2))) float v2f;
typedef __attribute__((ext_vector_type(8))) float v8f;

__device__ __forceinline__ v8f wmma4(v2f a, v2f b, v8f c) {
  // V_WMMA_F32_16X16X4_F32 : D = A(16x4) * B(4x16) + C, fp32 exact FMA chain
  return __builtin_amdgcn_wmma_f32_16x16x4_f32(false, a, false, b, (short)0, c,
                                               false, false);
}

// ---------------------------------------------------------------------------
// 1) channel mean+max pool, Hilbert scatter into feat[b][s*64+j][p]
// ---------------------------------------------------------------------------
__global__ void pool_gather_kernel(const float* __restrict__ x,
                                   const int* __restrict__ rehil,
                                   float* __restrict__ feat) {
  int pix = blockIdx.x * blockDim.x + threadIdx.x;   // 0..65535
  int b   = blockIdx.y;
  const float* xp = x + (size_t)b * CHN * NPIX + pix;
  float s = 0.0f, mx = -INFINITY;
#pragma unroll 4
  for (int c = 0; c < CHN; ++c) {
    float v = xp[(size_t)c * NPIX];
    s += v;
    mx = fmaxf(mx, v);
  }
  float mean = s * (1.0f / (float)CHN);
  int d = rehil[pix];                 // curve position of this pixel
  int p = d >> 6, j = d & 63;
  float* fb = feat + (size_t)b * CHN * PATHS;
  fb[(size_t)j * PATHS + p]          = mean;  // s=0 rows
  fb[(size_t)(PSQ + j) * PATHS + p]  = mx;    // s=1 rows
}

// ---------------------------------------------------------------------------
// 2) Q/K/V = W(128x128) @ feat(128x1024) + bias   (blockIdx.z selects which)
//    one wave -> one 16x16 output tile, K-loop of 32 x WMMA_F32_16X16X4_F32
// ---------------------------------------------------------------------------
__global__ void qkv_kernel(const float* __restrict__ feat,
                           const float* __restrict__ Wq, const float* __restrict__ bq,
                           const float* __restrict__ Wk, const float* __restrict__ bk,
                           const float* __restrict__ Wv, const float* __restrict__ bv,
                           float* __restrict__ Q, float* __restrict__ K,
                           float* __restrict__ V) {
  const float* W; const float* bias; float* Y;
  if (blockIdx.z == 0)      { W = Wq; bias = bq; Y = Q; }
  else if (blockIdx.z == 1) { W = Wk; bias = bk; Y = K; }
  else                      { W = Wv; bias = bv; Y = V; }

  int b     = blockIdx.y;
  int tile  = blockIdx.x * 8 + (threadIdx.x >> 5);  // 0..511
  int oBase = (tile >> 6) * 16;                     // 8 o-tiles
  int pBase = (tile & 63) * 16;                     // 64 p-tiles
  int lane  = threadIdx.x & 31, lo = lane & 15, hi = lane >> 4;

  const float* fb = feat + (size_t)b * CHN * PATHS;
  v8f acc = {};
  for (int k = 0; k < CHN; k += 4) {
    int k0 = k + 2 * hi;
    v2f a, bb;
    a[0]  = W[(size_t)(oBase + lo) * CHN + k0];
    a[1]  = W[(size_t)(oBase + lo) * CHN + k0 + 1];
    bb[0] = fb[(size_t)k0 * PATHS + pBase + lo];
    bb[1] = fb[(size_t)(k0 + 1) * PATHS + pBase + lo];
    acc = wmma4(a, bb, acc);
  }
  float* Yb = Y + (size_t)b * CHN * PATHS;
#pragma unroll
  for (int r = 0; r < 8; ++r) {
    int o = oBase + r + 8 * hi;
    Yb[(size_t)o * PATHS + pBase + lo] = acc[r] + bias[o];
  }
}

// ---------------------------------------------------------------------------
// 3) ST[q][p] = sum_c K[c][q] * Q[c][p]   (== scores[p][q], stored q-major so
//    the softmax over p is a contiguous-row reduction)
// ---------------------------------------------------------------------------
__global__ void scores_kernel(const float* __restrict__ Q,
                              const float* __restrict__ K,
                              float* __restrict__ ST) {
  int b     = blockIdx.y;
  int tile  = blockIdx.x * 8 + (threadIdx.x >> 5);  // 0..4095
  int qBase = (tile >> 6) * 16;
  int pBase = (tile & 63) * 16;
  int lane  = threadIdx.x & 31, lo = lane & 15, hi = lane >> 4;

  const float* Qb = Q + (size_t)b * CHN * PATHS;
  const float* Kb = K + (size_t)b * CHN * PATHS;
  v8f acc = {};
  for (int c = 0; c < CHN; c += 4) {
    int c0 = c + 2 * hi;
    v2f a, bb;
    a[0]  = Kb[(size_t)c0 * PATHS + qBase + lo];
    a[1]  = Kb[(size_t)(c0 + 1) * PATHS + qBase + lo];
    bb[0] = Qb[(size_t)c0 * PATHS + pBase + lo];
    bb[1] = Qb[(size_t)(c0 + 1) * PATHS + pBase + lo];
    acc = wmma4(a, bb, acc);
  }
  float* Sb = ST + (size_t)b * PATHS * PATHS;
#pragma unroll
  for (int r = 0; r < 8; ++r)
    Sb[(size_t)(qBase + r + 8 * hi) * PATHS + pBase + lo] = acc[r];
}

// ---------------------------------------------------------------------------
// 4) softmax over p (contiguous) per row q; one 256-thread block per row
// ---------------------------------------------------------------------------
__global__ void softmax_kernel(float* __restrict__ ST) {
  float* row = ST + (size_t)blockIdx.x * PATHS;
  int t = threadIdx.x;
  __shared__ float red[256];
  float v0 = row[t], v1 = row[t + 256], v2 = row[t + 512], v3 = row[t + 768];
  float m = fmaxf(fmaxf(v0, v1), fmaxf(v2, v3));
  red[t] = m; __syncthreads();
  for (int s = 128; s > 0; s >>= 1) {
    if (t < s) red[t] = fmaxf(red[t], red[t + s]);
    __syncthreads();
  }
  m = red[0]; __syncthreads();
  v0 = expf(v0 - m); v1 = expf(v1 - m); v2 = expf(v2 - m); v3 = expf(v3 - m);
  red[t] = v0 + v1 + v2 + v3; __syncthreads();
  for (int s = 128; s > 0; s >>= 1) {
    if (t < s) red[t] += red[t + s];
    __syncthreads();
  }
  float inv = 1.0f / red[0];
  row[t] = v0 * inv; row[t + 256] = v1 * inv;
  row[t + 512] = v2 * inv; row[t + 768] = v3 * inv;
}

// ---------------------------------------------------------------------------
// 5) x3c[c][p] = sum_q V[c][q] * ST[q][p]
// ---------------------------------------------------------------------------
__global__ void av_kernel(const float* __restrict__ ST,
                          const float* __restrict__ V,
                          float* __restrict__ x3c) {
  int b     = blockIdx.y;
  int tile  = blockIdx.x * 8 + (threadIdx.x >> 5);  // 0..511
  int cBase = (tile >> 6) * 16;
  int pBase = (tile & 63) * 16;
  int lane  = threadIdx.x & 31, lo = lane & 15, hi = lane >> 4;

  const float* Vb = V + (size_t)b * CHN * PATHS;
  const float* Sb = ST + (size_t)b * PATHS * PATHS;
  v8f acc = {};
  for (int q = 0; q < PATHS; q += 4) {
    int q0 = q + 2 * hi;
    v2f a, bb;
    a[0]  = Vb[(size_t)(cBase + lo) * PATHS + q0];
    a[1]  = Vb[(size_t)(cBase + lo) * PATHS + q0 + 1];
    bb[0] = Sb[(size_t)q0 * PATHS + pBase + lo];
    bb[1] = Sb[(size_t)(q0 + 1) * PATHS + pBase + lo];
    acc = wmma4(a, bb, acc);
  }
  float* Xb = x3c + (size_t)b * CHN * PATHS;
#pragma unroll
  for (int r = 0; r < 8; ++r)
    Xb[(size_t)(cBase + r + 8 * hi) * PATHS + pBase + lo] = acc[r];
}

// ---------------------------------------------------------------------------
// 6) gate[b][pix] = sigmoid(Wpre0*restored0 + Wpre1*restored1 + bpre)
// ---------------------------------------------------------------------------
__global__ void gate_kernel(const float* __restrict__ x3c,
                            const int* __restrict__ rehil,
                            const float* __restrict__ Wpre,
                            const float* __restrict__ bpre,
                            float* __restrict__ gate) {
  int pix = blockIdx.x * blockDim.x + threadIdx.x;
  int b   = blockIdx.y;
  int d = rehil[pix];
  int p = d >> 6, j = d & 63;
  const float* Xb = x3c + (size_t)b * CHN * PATHS;
  float r0 = Xb[(size_t)j * PATHS + p];
  float r1 = Xb[(size_t)(PSQ + j) * PATHS + p];
  float z = Wpre[0] * r0 + Wpre[1] * r1 + bpre[0];
  gate[(size_t)b * NPIX + pix] = 1.0f / (1.0f + expf(-z));
}

// ---------------------------------------------------------------------------
// 7) out = gate * x   (bandwidth-critical: float4 loads/stores)
// ---------------------------------------------------------------------------
__global__ void apply_kernel(const float* __restrict__ x,
                             const float* __restrict__ gate,
                             float* __restrict__ out) {
  size_t i  = (size_t)blockIdx.x * blockDim.x + threadIdx.x;  // float4 index
  size_t fi = i * 4;
  size_t b   = fi >> 23;            // / (128*65536)
  size_t pix = fi & (NPIX - 1);
  float4 xv = ((const float4*)x)[i];
  float4 g  = ((const float4*)gate)[((size_t)b * NPIX + pix) >> 2];
  float4 o;
  o.x = xv.x * g.x; o.y = xv.y * g.y; o.z = xv.z * g.z; o.w = xv.w * g.w;
  ((float4*)out)[i] = o;
}

// ---------------------------------------------------------------------------
extern "C" void kernel_launch(void* const* d_in, const int* in_sizes, int n_in,
                              void* d_out, int out_size, void* d_ws, size_t ws_size,
                              hipStream_t stream) {
  const float* x     = (const float*)d_in[0];
  const float* Wq    = (const float*)d_in[1];
  const float* bq    = (const float*)d_in[2];
  const float* Wk    = (const float*)d_in[3];
  const float* bk    = (const float*)d_in[4];
  const float* Wv    = (const float*)d_in[5];
  const float* bv    = (const float*)d_in[6];
  const float* Wpre  = (const float*)d_in[7];
  const float* bpre  = (const float*)d_in[8];
  const int*   rehil = (const int*)d_in[10];

  const int B = in_sizes[0] / (CHN * NPIX);   // 8

  // workspace carve-up (floats)
  float* feat = (float*)d_ws;                         // B*128*1024
  float* Q    = feat + (size_t)B * CHN * PATHS;
  float* K    = Q    + (size_t)B * CHN * PATHS;
  float* V    = K    + (size_t)B * CHN * PATHS;
  float* ST   = V    + (size_t)B * CHN * PATHS;       // B*1024*1024
  float* x3c  = ST   + (size_t)B * PATHS * PATHS;
  float* gate = x3c  + (size_t)B * CHN * PATHS;

  // 1) pool + hilbert scatter
  pool_gather_kernel<<<dim3(NPIX / 256, B), 256, 0, stream>>>(x, rehil, feat);

  // 2) Q/K/V projections (WMMA): 512 tiles / batch / matrix, 8 waves per block
  qkv_kernel<<<dim3(64, B, 3), 256, 0, stream>>>(feat, Wq, bq, Wk, bk, Wv, bv,
                                                 Q, K, V);

  // 3) scores, stored transposed (WMMA): 4096 tiles / batch
  scores_kernel<<<dim3(512, B), 256, 0, stream>>>(Q, K, ST);

  // 4) softmax over p (per (b,q) row)
  softmax_kernel<<<dim3(B * PATHS), 256, 0, stream>>>(ST);

  // 5) attention * V (WMMA)
  av_kernel<<<dim3(64, B), 256, 0, stream>>>(ST, V, x3c);

  // 6) sigmoid gate per pixel
  gate_kernel<<<dim3(NPIX / 256, B), 256, 0, stream>>>(x3c, rehil, Wpre, bpre,
                                                       gate);

  // 7) elementwise apply (memory-bound pass)
  size_t total4 = (size_t)B * CHN * NPIX / 4;
  apply_kernel<<<dim3(total4 / 256), 256, 0, stream>>>(x, gate, (float*)d_out);
}